// MultiScaleSignatureEncoder_63410897158629
// MI455X (gfx1250) — compile-verified
//
#include <hip/hip_runtime.h>

#define B_N 8192
#define T_N 64
#define CIN 64
#define P_N 32
#define SIG_DIM 4160
#define OUT_STRIDE (3 * SIG_DIM)
#define WAVES_PER_WG 4
#define XPITCH 68    // 28 rows x 64ch, padded: stride-4 banks -> conflict-free column reads
#define HPITCH 36    // h rows of 32ch, padded
#define XROWS 28
#define HROWS 29     // 29*36*2 = 2088 floats -> doubles as 2048-float bounce buffer
#define XW_FLOATS (XROWS * XPITCH)              // 1904
#define H_FLOATS  (HROWS * HPITCH)              // 1044
#define WSLOT (XW_FLOATS + 2 * H_FLOATS)        // 3992 floats / wave slot (63872 B total)

typedef float v2f __attribute__((ext_vector_type(2)));
typedef float v8f __attribute__((ext_vector_type(8)));

__device__ __forceinline__ v8f wmma4(v2f a, v2f b, v8f c) {
  // D = A(16x4) * B(4x16) + C, full f32
  return __builtin_amdgcn_wmma_f32_16x16x4_f32(false, a, false, b, (short)0, c,
                                               false, false);
}

__device__ __forceinline__ float gelu_exact(float v) {
  return 0.5f * v * (1.0f + erff(v * 0.70710678118654752f));
}

__device__ __forceinline__ float clip50(float v) {
  return fminf(50.0f, fmaxf(-50.0f, v));
}

__launch_bounds__(WAVES_PER_WG * 32)
__global__ void msig_encoder_kernel(const float* __restrict__ x,
                                    const float* __restrict__ W1,
                                    const float* __restrict__ b1,
                                    const float* __restrict__ W2,
                                    const float* __restrict__ b2,
                                    const float* __restrict__ gamma,
                                    const float* __restrict__ beta,
                                    float* __restrict__ out) {
  __shared__ float smem[WAVES_PER_WG * WSLOT];

  const int wave = threadIdx.x >> 5;
  const int lane = threadIdx.x & 31;
  const int b = blockIdx.x * WAVES_PER_WG + wave;   // one wave == one batch element

  float* xw = smem + wave * WSLOT;
  float* h1 = xw + XW_FLOATS;
  float* h2 = h1 + H_FLOATS;

  const int lhalf = lane >> 4;       // which 16-lane half
  const int l16 = lane & 15;
  const int khalf = lhalf << 1;      // K offset of this half within a 4-wide chunk

  // ---- stage x rows 36..63 (union of all windows) into LDS, coalesced b128 ----
  const float* xb = x + (size_t)b * (T_N * CIN) + 36 * CIN;
#pragma unroll
  for (int q = 0; q < 14; ++q) {
    const int idx4 = q * 32 + lane;      // 448 float4's total
    const int flat = idx4 << 2;
    const int row = flat >> 6;
    const int ch = flat & 63;
    const float4 v = *(const float4*)(xb + flat);
    *(float4*)(xw + row * XPITCH + ch) = v;
  }

  const int WIN[3] = {7, 14, 28};

  for (int wi = 0; wi < 3; ++wi) {
    const int w = WIN[wi];
    const int Mt = (w + 15) >> 4;        // row tiles (1,1,2)
    const int rowOff = XROWS - w;        // window = last w of the 28 staged rows
    const float* W1w = W1 + wi * (CIN * P_N);
    const float* W2w = W2 + wi * (P_N * P_N);
    const float* b1w = b1 + wi * P_N;
    const float* b2w = b2 + wi * P_N;
    const int n0 = l16;
    const int n1 = 16 + l16;

    // ====== GEMM1: h1 = GELU(win @ W1 + b1)  (w x 64)·(64 x 32), 2 acc ILP ==
    for (int mt = 0; mt < Mt; ++mt) {
      int arow = rowOff + mt * 16 + l16;
      if (arow > XROWS - 1) arow = XROWS - 1;       // clamp padded rows
      const float* aBase = xw + arow * XPITCH;
      v8f acc0 = {}, acc1 = {};
#pragma unroll
      for (int kc = 0; kc < 16; ++kc) {
        const int k0 = kc * 4 + khalf;
        const v2f a = *(const v2f*)(aBase + k0);    // ds_read_b64
        v2f bb0, bb1;
        bb0.x = W1w[k0 * P_N + n0];
        bb0.y = W1w[(k0 + 1) * P_N + n0];
        bb1.x = W1w[k0 * P_N + n1];
        bb1.y = W1w[(k0 + 1) * P_N + n1];
        acc0 = wmma4(a, bb0, acc0);
        acc1 = wmma4(a, bb1, acc1);
      }
      const float bias0 = b1w[n0];
      const float bias1 = b1w[n1];
#pragma unroll
      for (int r = 0; r < 8; ++r) {
        const int m = mt * 16 + lhalf * 8 + r;      // C layout row
        if (m < w) {
          h1[m * HPITCH + n0] = gelu_exact(acc0[r] + bias0);
          h1[m * HPITCH + n1] = gelu_exact(acc1[r] + bias1);
        }
      }
    }

    // ====== GEMM2: h2 = h1 @ W2 + b2  (w x 32)·(32 x 32), 2 acc ILP =========
    for (int mt = 0; mt < Mt; ++mt) {
      int arow = mt * 16 + l16;
      if (arow > w - 1) arow = w - 1;
      const float* aBase = h1 + arow * HPITCH;
      v8f acc0 = {}, acc1 = {};
#pragma unroll
      for (int kc = 0; kc < 8; ++kc) {
        const int k0 = kc * 4 + khalf;
        const v2f a = *(const v2f*)(aBase + k0);
        v2f bb0, bb1;
        bb0.x = W2w[k0 * P_N + n0];
        bb0.y = W2w[(k0 + 1) * P_N + n0];
        bb1.x = W2w[k0 * P_N + n1];
        bb1.y = W2w[(k0 + 1) * P_N + n1];
        acc0 = wmma4(a, bb0, acc0);
        acc1 = wmma4(a, bb1, acc1);
      }
      const float bias0 = b2w[n0];
      const float bias1 = b2w[n1];
#pragma unroll
      for (int r = 0; r < 8; ++r) {
        const int m = mt * 16 + lhalf * 8 + r;
        if (m < w) {
          h2[m * HPITCH + n0] = acc0[r] + bias0;
          h2[m * HPITCH + n1] = acc1[r] + bias1;
        }
      }
    }

    // ======== level2 = Inc^T(64xK) @ Prev(Kx64), K = 2w-2, 16 WMMA tiles ====
    // inc[2k]=(dh[k],0) inc[2k+1]=(0,dh[k]); prev[2k]=(hk-h0,hk-h0),
    // prev[2k+1]=(h{k+1}-h0, hk-h0); both VGPR slots of a chunk share k.
    const float h0a = h2[l16];
    const float h0b = h2[16 + l16];
    const int K = 2 * w - 2;
    const int nkc = (K + 3) >> 2;

    v8f acc2[16];
#pragma unroll
    for (int t = 0; t < 16; ++t) acc2[t] = (v8f){};

    for (int kc = 0; kc < nkc; ++kc) {
      const int tbase = kc * 4 + khalf;          // even; t = tbase + v
      const bool valid = tbase < K;              // K even -> both slots same validity
      int kk = tbase >> 1;
      if (kk > XROWS - 2) kk = XROWS - 2;        // keep LDS reads in-bounds on padding
      const float p0a = h2[kk * HPITCH + l16];
      const float p0b = h2[kk * HPITCH + 16 + l16];
      const float p1a = h2[(kk + 1) * HPITCH + l16];
      const float p1b = h2[(kk + 1) * HPITCH + 16 + l16];
      const float da = valid ? (p1a - p0a) : 0.0f;   // dh[k]
      const float db = valid ? (p1b - p0b) : 0.0f;
      const float q0a = valid ? (p0a - h0a) : 0.0f;  // h[k]-h0
      const float q0b = valid ? (p0b - h0b) : 0.0f;
      const float q1a = valid ? (p1a - h0a) : 0.0f;  // h[k+1]-h0
      const float q1b = valid ? (p1b - h0b) : 0.0f;

      const v2f a0 = {da, 0.0f};    // i in [0,16)   : even t only
      const v2f a1 = {db, 0.0f};    // i in [16,32)
      const v2f a2 = {0.0f, da};    // i in [32,48)  : odd t only
      const v2f a3 = {0.0f, db};    // i in [48,64)
      const v2f bb0 = {q0a, q1a};   // j in [0,16)
      const v2f bb1 = {q0b, q1b};   // j in [16,32)
      const v2f bb2 = {q0a, q0a};   // j in [32,48)
      const v2f bb3 = {q0b, q0b};   // j in [48,64)

      acc2[0]  = wmma4(a0, bb0, acc2[0]);
      acc2[1]  = wmma4(a0, bb1, acc2[1]);
      acc2[2]  = wmma4(a0, bb2, acc2[2]);
      acc2[3]  = wmma4(a0, bb3, acc2[3]);
      acc2[4]  = wmma4(a1, bb0, acc2[4]);
      acc2[5]  = wmma4(a1, bb1, acc2[5]);
      acc2[6]  = wmma4(a1, bb2, acc2[6]);
      acc2[7]  = wmma4(a1, bb3, acc2[7]);
      acc2[8]  = wmma4(a2, bb0, acc2[8]);
      acc2[9]  = wmma4(a2, bb1, acc2[9]);
      acc2[10] = wmma4(a2, bb2, acc2[10]);
      acc2[11] = wmma4(a2, bb3, acc2[11]);
      acc2[12] = wmma4(a3, bb0, acc2[12]);
      acc2[13] = wmma4(a3, bb1, acc2[13]);
      acc2[14] = wmma4(a3, bb2, acc2[14]);
      acc2[15] = wmma4(a3, bb3, acc2[15]);
    }

    // ---- level1 (duplicated halves), clip, wave-wide mean/var ----
    float l1 = clip50(h2[(w - 1) * HPITCH + lane] - h2[lane]);
    float s1 = 2.0f * l1;
    float s2 = 2.0f * l1 * l1;
#pragma unroll
    for (int t = 0; t < 16; ++t) {
#pragma unroll
      for (int r = 0; r < 8; ++r) {
        float v = clip50(acc2[t][r]);
        acc2[t][r] = v;
        s1 += v;
        s2 += v * v;
      }
    }
#pragma unroll
    for (int off = 16; off > 0; off >>= 1) {
      s1 += __shfl_xor(s1, off, 32);
      s2 += __shfl_xor(s2, off, 32);
    }
    const float invN = 1.0f / (float)SIG_DIM;
    const float mean = s1 * invN;
    const float var = fmaxf(s2 * invN - mean * mean, 0.0f);
    const float rstd = rsqrtf(var + 1e-5f);

    // ---- LayerNorm + write ----
    float* outp = out + (size_t)b * OUT_STRIDE + wi * SIG_DIM;
    const float* gw = gamma + wi * SIG_DIM;
    const float* bw = beta + wi * SIG_DIM;
    {
      const float vn = (l1 - mean) * rstd;
      outp[lane] = vn * gw[lane] + bw[lane];
      outp[32 + lane] = vn * gw[32 + lane] + bw[32 + lane];
    }

    // level2: bounce 2048-float chunks through LDS (h1+h2 region is dead now)
    // so gamma/beta loads and output stores are fully-coalesced b128.
    float* bounce = h1;   // 2088 floats available, use 2048
#pragma unroll
    for (int c = 0; c < 2; ++c) {
      // deposit rows i = c*32 .. c*32+31 (tiles ti = 2c, 2c+1)
#pragma unroll
      for (int tt = 0; tt < 2; ++tt) {
        const int ti = c * 2 + tt;
#pragma unroll
        for (int tj = 0; tj < 4; ++tj) {
#pragma unroll
          for (int r = 0; r < 8; ++r) {
            const int il = tt * 16 + lhalf * 8 + r;    // local row 0..31
            bounce[il * 64 + tj * 16 + l16] = acc2[ti * 4 + tj][r];
          }
        }
      }
      // linear re-read + normalize + coalesced store (512B per instruction)
#pragma unroll
      for (int q = 0; q < 16; ++q) {
        const int flat = (q * 32 + lane) << 2;         // 0..2044
        const float4 v = *(const float4*)(bounce + flat);
        const int idx = 64 + c * 2048 + flat;
        const float4 g = *(const float4*)(gw + idx);
        const float4 bt = *(const float4*)(bw + idx);
        float4 o;
        o.x = (v.x - mean) * rstd * g.x + bt.x;
        o.y = (v.y - mean) * rstd * g.y + bt.y;
        o.z = (v.z - mean) * rstd * g.z + bt.z;
        o.w = (v.w - mean) * rstd * g.w + bt.w;
        *(float4*)(outp + idx) = o;
      }
    }
  }
}

extern "C" void kernel_launch(void* const* d_in, const int* in_sizes, int n_in,
                              void* d_out, int out_size, void* d_ws, size_t ws_size,
                              hipStream_t stream) {
  const float* x = (const float*)d_in[0];
  const float* W1 = (const float*)d_in[1];
  const float* b1 = (const float*)d_in[2];
  const float* W2 = (const float*)d_in[3];
  const float* b2 = (const float*)d_in[4];
  const float* gamma = (const float*)d_in[5];
  const float* beta = (const float*)d_in[6];
  float* out = (float*)d_out;

  dim3 grid(B_N / WAVES_PER_WG);   // 2048 workgroups
  dim3 block(WAVES_PER_WG * 32);   // 128 threads = 4 wave32
  msig_encoder_kernel<<<grid, block, 0, stream>>>(x, W1, b1, W2, b2, gamma, beta, out);
}